// ExponentialEnvelopes_21895743275104
// MI455X (gfx1250) — compile-verified
//
#include <hip/hip_runtime.h>
#include <hip/hip_bf16.h>
#include <math.h>

// ---------------------------------------------------------------------------
// ExponentialEnvelopes for MI455X (gfx1250, wave32)
//
// out[s,d,e,j] = sm[e,j] ? sum_n pi_s[d*128+j,n] * exp(-|dist_s[e,n]*zeta_s[d*128+j,n]|)
//                        : (e==j ? 1 : 0)
//
// Pipeline (all on `stream`):
//   1) feats_kernel      : masked mean-center nuclei -> feats[128,4]
//   2) gemm4_wmma_kernel : zetaT/piT = W^T x feats^T via v_wmma_f32_16x16x4_f32
//                          (zeta pre-scaled by log2(e) so the hot loop uses
//                           v_exp_f32 with no extra multiply)
//   3) dist_kernel       : masked-safe electron-nucleus distances
//   4) envelope_kernel   : TRANS-bound exp2/accumulate; operand tiles streamed
//                          into LDS with double-buffered
//                          global_load_async_to_lds_b128 (ASYNCcnt)
// ---------------------------------------------------------------------------

typedef __attribute__((ext_vector_type(2))) float v2f;
typedef __attribute__((ext_vector_type(8))) float v8f;

#define ME   128
#define MN   128
#define NDET 32
#define NORB (NDET * ME)   // 4096
#define JPAD 132           // row stride in floats; %64 == 4 -> lane j maps to
                           // bank (4*j + n) % 64: conflict-free for 16 lanes/
                           // cycle, and 16B-aligned rows for B128 async loads
#define LOG2E 1.44269504088896340736f

// ---------------------------------------------------------------------------
// CDNA5 async global->LDS copy (bypasses VGPRs, tracked by ASYNCcnt).
// l must be 16B-aligned LDS, g 16B-aligned global.
// ---------------------------------------------------------------------------
__device__ __forceinline__ void async_copy_b128(const float* g, float* l) {
  unsigned lds_off = (unsigned)(size_t)l;           // low 32 bits = LDS offset
  unsigned long long ga = (unsigned long long)(size_t)g;
  asm volatile("global_load_async_to_lds_b128 %0, %1, off"
               :: "v"(lds_off), "v"(ga) : "memory");
}
__device__ __forceinline__ void wait_async_le8() {
  asm volatile("s_wait_asynccnt 0x8" ::: "memory");
}
__device__ __forceinline__ void wait_async_0() {
  asm volatile("s_wait_asynccnt 0x0" ::: "memory");
}

// ---------------------------------------------------------------------------
// 1) Nuclear features: feats[n] = mask * [charge, coords - masked_mean]
// ---------------------------------------------------------------------------
__global__ __launch_bounds__(128) void feats_kernel(
    const float* __restrict__ nuc_coords,
    const float* __restrict__ nuc_charges,
    const unsigned char* __restrict__ nuc_mask,
    float* __restrict__ feats) {
  __shared__ float red[4][MN];
  int n = threadIdx.x;                       // 0..127
  float m = nuc_mask[n] ? 1.0f : 0.0f;
  float x = nuc_coords[n * 3 + 0];
  float y = nuc_coords[n * 3 + 1];
  float z = nuc_coords[n * 3 + 2];
  red[0][n] = m * x; red[1][n] = m * y; red[2][n] = m * z; red[3][n] = m;
  __syncthreads();
  for (int s = 64; s > 0; s >>= 1) {
    if (n < s) {
      red[0][n] += red[0][n + s];
      red[1][n] += red[1][n + s];
      red[2][n] += red[2][n + s];
      red[3][n] += red[3][n + s];
    }
    __syncthreads();
  }
  float denom = fmaxf(red[3][0], 1.0f);
  float mx = red[0][0] / denom, my = red[1][0] / denom, mz = red[2][0] / denom;
  feats[n * 4 + 0] = m * nuc_charges[n];
  feats[n * 4 + 1] = m * (x - mx);
  feats[n * 4 + 2] = m * (y - my);
  feats[n * 4 + 3] = m * (z - mz);
}

// ---------------------------------------------------------------------------
// 2) zetaT[o][n] = log2e * sum_k W_pi[k][o] * feats[n][k]   (mat 0, 2)
//    piT  [o][n] =         sum_k W_ze[k][o] * feats[n][k]   (mat 1, 3)
//    One v_wmma_f32_16x16x4_f32 per 16x16 tile: D = A(16x4) x B(4x16).
//    VGPR layouts per CDNA5 ISA 7.12.2 (A lanes16-31 carry K={2,3}; D VGPR r
//    holds rows r / r+8). 8192 waves = 4 matrices x 256 o-tiles x 8 n-tiles.
// ---------------------------------------------------------------------------
__global__ __launch_bounds__(256) void gemm4_wmma_kernel(
    const float* __restrict__ feats,
    const float* __restrict__ Wpi_up, const float* __restrict__ Wze_up,
    const float* __restrict__ Wpi_dn, const float* __restrict__ Wze_dn,
    float* __restrict__ zpu, float* __restrict__ ppu,
    float* __restrict__ zpd, float* __restrict__ ppd) {
  int wave = threadIdx.x >> 5;
  int lane = threadIdx.x & 31;
  int gw   = blockIdx.x * 8 + wave;          // 0..8191 (wave-uniform)
  int mat  = gw >> 11;                       // 0..3
  int t    = gw & 2047;
  int o0   = (t >> 3) << 4;                  // orbital tile base
  int n0   = (t & 7) << 4;                   // nucleus tile base

  const float* W   = (mat == 0) ? Wpi_up : (mat == 1) ? Wze_up
                   : (mat == 2) ? Wpi_dn : Wze_dn;
  float*       Dst = (mat == 0) ? zpu : (mat == 1) ? ppu
                   : (mat == 2) ? zpd : ppd;
  float scale = ((mat & 1) == 0) ? LOG2E : 1.0f;   // fold log2(e) into zeta

  int half = lane >> 4;                      // 0: K={0,1}, 1: K={2,3}
  int l    = lane & 15;

  v2f a, b;
  a.x = W[(half * 2 + 0) * NORB + o0 + l];   // A[l][2h+0] = W[k][o0+l]
  a.y = W[(half * 2 + 1) * NORB + o0 + l];
  b.x = feats[(n0 + l) * 4 + (half * 2 + 0)];// B[2h+0][l] = feats[n0+l][k]
  b.y = feats[(n0 + l) * 4 + (half * 2 + 1)];

  v8f c = {};
  // 8 args: (neg_a, A, neg_b, B, c_mod, C, reuse_a, reuse_b)
  c = __builtin_amdgcn_wmma_f32_16x16x4_f32(false, a, false, b, (short)0, c,
                                            false, false);
#pragma unroll
  for (int r = 0; r < 8; ++r) {
    Dst[(o0 + r + half * 8) * MN + (n0 + l)] = c[r] * scale;
  }
}

// ---------------------------------------------------------------------------
// 3) dist[s][e][n] = sqrt(mask ? |e-n|^2 : 1.0)   (safe sqrt per reference)
// ---------------------------------------------------------------------------
__global__ __launch_bounds__(256) void dist_kernel(
    const float* __restrict__ up, const float* __restrict__ dn,
    const float* __restrict__ nuc,
    const unsigned char* __restrict__ um, const unsigned char* __restrict__ dm,
    const unsigned char* __restrict__ nm,
    float* __restrict__ dist) {
  int f = blockIdx.x * blockDim.x + threadIdx.x;   // < 2*128*128
  int s = f >> 14;
  int e = (f >> 7) & 127;
  int n = f & 127;
  const float* ec = s ? dn : up;
  const unsigned char* em = s ? dm : um;
  float dx = ec[e * 3 + 0] - nuc[n * 3 + 0];
  float dy = ec[e * 3 + 1] - nuc[n * 3 + 1];
  float dz = ec[e * 3 + 2] - nuc[n * 3 + 2];
  float d2 = dx * dx + dy * dy + dz * dz;
  bool msk = (em[e] != 0) && (nm[n] != 0);
  dist[f] = sqrtf(msk ? d2 : 1.0f);
}

// ---------------------------------------------------------------------------
// 4) Envelope accumulation. Block = (spin, det, e-tile of 32). 256 threads:
//    lane = orbital j within 32-chunk, wave id = electron group of 4.
//    zeta/pi tiles double-buffered via async-to-LDS; 2048 v_exp_f32/thread.
// ---------------------------------------------------------------------------
__global__ __launch_bounds__(256) void envelope_kernel(
    const float* __restrict__ zetaT_up, const float* __restrict__ piT_up,
    const float* __restrict__ zetaT_dn, const float* __restrict__ piT_dn,
    const float* __restrict__ dist,
    const unsigned char* __restrict__ um, const unsigned char* __restrict__ dm,
    float* __restrict__ out) {
  __shared__ float s_dist[32][JPAD];
  __shared__ float s_zeta[2][32][JPAD];
  __shared__ float s_pi[2][32][JPAD];

  int bid = blockIdx.x;
  int s   = bid >> 7;           // spin
  int d   = (bid >> 2) & 31;    // determinant
  int e0  = (bid & 3) * 32;     // electron tile base

  const float* zT = s ? zetaT_dn : zetaT_up;
  const float* pT = s ? piT_dn : piT_up;
  const unsigned char* em = s ? dm : um;

  int tid = threadIdx.x;
  // stage distance tile [32 e][128 n] (plain path; read once)
  for (int i = tid; i < 32 * 128; i += 256) {
    int r = i >> 7, cc = i & 127;
    s_dist[r][cc] = dist[s * ME * MN + (e0 + r) * MN + cc];
  }

  const float* zbase = zT + (size_t)(d * ME) * MN;   // 128 rows of 128
  const float* pbase = pT + (size_t)(d * ME) * MN;

  // issue async loads for one 32-row chunk into buffer `buf`
  // (4 B128 per thread per array = 8 outstanding per thread per chunk)
  auto issue_chunk = [&](int jc, int buf) {
#pragma unroll
    for (int k = 0; k < 4; ++k) {
      int f  = tid + k * 256;        // float4 index, < 1024
      int r  = f >> 5;               // row 0..31
      int c4 = (f & 31) * 4;         // col 0..124 step 4
      const float* gz = zbase + (size_t)(jc * 32 + r) * MN + c4;
      const float* gp = pbase + (size_t)(jc * 32 + r) * MN + c4;
      async_copy_b128(gz, &s_zeta[buf][r][c4]);
      async_copy_b128(gp, &s_pi[buf][r][c4]);
    }
  };

  issue_chunk(0, 0);                 // prime the pipeline

  int jj = tid & 31;                 // orbital lane within chunk
  int eg = tid >> 5;                 // electron group (wave-uniform)
  int ee = eg * 4;

  for (int jc = 0; jc < 4; ++jc) {
    int buf = jc & 1;
    if (jc < 3) {
      issue_chunk(jc + 1, buf ^ 1);  // overlap next chunk with compute
      wait_async_le8();              // retire only chunk jc's 8 loads
    } else {
      wait_async_0();
    }
    __syncthreads();                 // chunk jc visible to all waves
                                     // (also covers s_dist on jc==0)

    float acc0 = 0.f, acc1 = 0.f, acc2 = 0.f, acc3 = 0.f;
#pragma unroll 4
    for (int n = 0; n < MN; ++n) {
      float z = s_zeta[buf][jj][n];  // pre-scaled by log2(e)
      float p = s_pi[buf][jj][n];
      float t0 = s_dist[ee + 0][n] * z;
      float t1 = s_dist[ee + 1][n] * z;
      float t2 = s_dist[ee + 2][n] * z;
      float t3 = s_dist[ee + 3][n] * z;
      acc0 += p * __builtin_amdgcn_exp2f(-fabsf(t0));   // v_exp_f32, no mul
      acc1 += p * __builtin_amdgcn_exp2f(-fabsf(t1));
      acc2 += p * __builtin_amdgcn_exp2f(-fabsf(t2));
      acc3 += p * __builtin_amdgcn_exp2f(-fabsf(t3));
    }

    int jg = jc * 32 + jj;
    bool mj = (em[jg] != 0);
    float accv[4] = {acc0, acc1, acc2, acc3};
#pragma unroll
    for (int u = 0; u < 4; ++u) {
      int eglob = e0 + ee + u;
      bool sm = (em[eglob] != 0) && mj;
      float v = sm ? accv[u] : (eglob == jg ? 1.0f : 0.0f);
      out[(((size_t)(s * NDET + d)) * ME + eglob) * ME + jg] = v;
    }
    __syncthreads();                 // all reads of buf done before it is
                                     // overwritten by iteration jc+1's issue
  }
}

// ---------------------------------------------------------------------------
extern "C" void kernel_launch(void* const* d_in, const int* in_sizes, int n_in,
                              void* d_out, int out_size, void* d_ws, size_t ws_size,
                              hipStream_t stream) {
  const float* up   = (const float*)d_in[0];
  const float* dn   = (const float*)d_in[1];
  const float* nuc  = (const float*)d_in[2];
  const float* chg  = (const float*)d_in[3];
  const float* Wpu  = (const float*)d_in[4];
  const float* Wzu  = (const float*)d_in[5];
  const float* Wpd  = (const float*)d_in[6];
  const float* Wzd  = (const float*)d_in[7];
  const unsigned char* um = (const unsigned char*)d_in[8];
  const unsigned char* dm = (const unsigned char*)d_in[9];
  const unsigned char* nm = (const unsigned char*)d_in[10];

  // workspace layout (floats): feats | zetaT_up | piT_up | zetaT_dn | piT_dn | dist
  float* ws    = (float*)d_ws;
  float* feats = ws;                       // 128*4
  float* zpu   = feats + MN * 4;           // 4096*128 each
  float* ppu   = zpu + (size_t)NORB * MN;
  float* zpd   = ppu + (size_t)NORB * MN;
  float* ppd   = zpd + (size_t)NORB * MN;
  float* dist  = ppd + (size_t)NORB * MN;  // 2*128*128

  feats_kernel<<<1, 128, 0, stream>>>(nuc, chg, nm, feats);
  gemm4_wmma_kernel<<<1024, 256, 0, stream>>>(feats, Wpu, Wzu, Wpd, Wzd,
                                              zpu, ppu, zpd, ppd);
  dist_kernel<<<128, 256, 0, stream>>>(up, dn, nuc, um, dm, nm, dist);
  envelope_kernel<<<256, 256, 0, stream>>>(zpu, ppu, zpd, ppd, dist, um, dm,
                                           (float*)d_out);
}